// TopKRouter_42099269436304
// MI455X (gfx1250) — compile-verified
//
#include <hip/hip_runtime.h>
#include <hip/hip_bf16.h>
#include <math.h>

// ---------------------------------------------------------------------------
// MoE top-2 router for MI455X (gfx1250, wave32).
// logits = A[B,D] x W[E,D]^T via V_WMMA_F32_16X16X4_F32 (full fp32: the op is
// HBM-bound at ~11.5us for 268MB of A, so fp32 matrix math costs nothing and
// preserves exact top-k ordering vs the fp32 reference).
// ---------------------------------------------------------------------------

typedef float v2f __attribute__((ext_vector_type(2)));
typedef float v8f __attribute__((ext_vector_type(8)));

#define B_DIM 16384
#define D_DIM 4096
#define E_DIM 64
#define TOPK  2

#define WAVES_PER_BLOCK 4
#define ROWS_PER_WAVE   16
#define ROWS_PER_BLOCK  (WAVES_PER_BLOCK * ROWS_PER_WAVE)
#define LDSTRIDE        65   // 64 + 1 pad: column reads hit distinct LDS banks

// Zero the global per-expert accumulators (must happen every launch: the
// harness poisons d_ws once and never re-poisons between graph replays).
__global__ void router_ws_zero(float* ws) {
    int t = threadIdx.x;
    if (t < 2 * E_DIM) ws[t] = 0.0f;
}

__global__ __launch_bounds__(WAVES_PER_BLOCK * 32)
void router_gemm_topk(const float* __restrict__ A, const float* __restrict__ W,
                      float* __restrict__ out, float* __restrict__ Psum,
                      float* __restrict__ Cnt) {
    __shared__ float lg[WAVES_PER_BLOCK][ROWS_PER_WAVE * LDSTRIDE];
    __shared__ float Ppart[E_DIM];
    __shared__ float Cpart[E_DIM];

    const int tid = threadIdx.x;
    for (int e = tid; e < E_DIM; e += blockDim.x) { Ppart[e] = 0.0f; Cpart[e] = 0.0f; }
    __syncthreads();

    const int wave = tid >> 5;
    const int lane = tid & 31;
    const int m    = lane & 15;   // row-in-tile (A) / N-in-tile (B)
    const int half = lane >> 4;   // K half-select per ISA 16x4 f32 layout
    const int row0 = (blockIdx.x * WAVES_PER_BLOCK + wave) * ROWS_PER_WAVE;

    // A fragment: lane holds A[row0+m][k + 2*half + {0,1}]  (one b64 load)
    const float* aptr = A + (size_t)(row0 + m) * D_DIM + 2 * half;
    // B fragment (B[k][n] = W[n][k]): lane holds W[t*16+m][k + 2*half + {0,1}]
    const float* b0 = W + (size_t)(0 * 16 + m) * D_DIM + 2 * half;
    const float* b1 = W + (size_t)(1 * 16 + m) * D_DIM + 2 * half;
    const float* b2 = W + (size_t)(2 * 16 + m) * D_DIM + 2 * half;
    const float* b3 = W + (size_t)(3 * 16 + m) * D_DIM + 2 * half;

    v8f acc0 = {}, acc1 = {}, acc2 = {}, acc3 = {};

    for (int k = 0; k < D_DIM; k += 16) {
        // speculative prefetch ~4KB ahead in this lane's A row
        __builtin_prefetch(aptr + k + 1024, 0, 0);
#pragma unroll
        for (int kk = 0; kk < 16; kk += 4) {
            const v2f a  = *(const v2f*)(aptr + k + kk);
            const v2f w0 = *(const v2f*)(b0 + k + kk);
            const v2f w1 = *(const v2f*)(b1 + k + kk);
            const v2f w2 = *(const v2f*)(b2 + k + kk);
            const v2f w3 = *(const v2f*)(b3 + k + kk);
            acc0 = __builtin_amdgcn_wmma_f32_16x16x4_f32(false, a, false, w0,
                                                         (short)0, acc0, false, false);
            acc1 = __builtin_amdgcn_wmma_f32_16x16x4_f32(false, a, false, w1,
                                                         (short)0, acc1, false, false);
            acc2 = __builtin_amdgcn_wmma_f32_16x16x4_f32(false, a, false, w2,
                                                         (short)0, acc2, false, false);
            acc3 = __builtin_amdgcn_wmma_f32_16x16x4_f32(false, a, false, w3,
                                                         (short)0, acc3, false, false);
        }
    }

    // Spill 16x64 logits tile to LDS. C/D layout: elem j of lane L is
    // (M = j + 8*(L>>4), N = L&15) within each 16-wide N tile.
    {
        float* base = lg[wave];
#pragma unroll
        for (int j = 0; j < 8; ++j) {
            const int r = j + 8 * half;
            base[r * LDSTRIDE +  0 + m] = acc0[j];
            base[r * LDSTRIDE + 16 + m] = acc1[j];
            base[r * LDSTRIDE + 32 + m] = acc2[j];
            base[r * LDSTRIDE + 48 + m] = acc3[j];
        }
    }
    __syncthreads();

    // Per-row softmax / top-2 on lanes 0..15 (one row each).
    if (lane < 16) {
        const float* lrow = lg[wave] + lane * LDSTRIDE;

        float v0 = -INFINITY, v1 = -INFINITY;
        int   i0 = 0, i1 = 0;
        for (int e = 0; e < E_DIM; ++e) {
            const float l = lrow[e];
            if (l > v0)      { v1 = v0; i1 = i0; v0 = l; i0 = e; }
            else if (l > v1) { v1 = l; i1 = e; }
        }

        float s = 0.0f;
        for (int e = 0; e < E_DIM; ++e) s += __expf(lrow[e] - v0);
        const float inv = 1.0f / s;
        for (int e = 0; e < E_DIM; ++e)
            atomicAdd(&Ppart[e], __expf(lrow[e] - v0) * inv);

        atomicAdd(&Cpart[i0], 1.0f);
        atomicAdd(&Cpart[i1], 1.0f);

        // top-2 softmax: w0 = 1/(1+exp(v1-v0)), w1 = exp(v1-v0)*w0
        const float e1 = __expf(v1 - v0);
        const float w0 = 1.0f / (1.0f + e1);
        const float w1 = e1 * w0;

        const int rg = row0 + lane;
        out[2 * rg + 0] = w0;
        out[2 * rg + 1] = w1;
        out[2 * B_DIM + 2 * rg + 0] = (float)i0;   // indices stored as float
        out[2 * B_DIM + 2 * rg + 1] = (float)i1;
    }
    __syncthreads();

    // One global atomic per expert per block.
    for (int e = tid; e < E_DIM; e += blockDim.x) {
        atomicAdd(&Psum[e], Ppart[e]);
        atomicAdd(&Cnt[e],  Cpart[e]);
    }
}

__global__ void router_finalize(const float* __restrict__ Psum,
                                const float* __restrict__ Cnt,
                                float* __restrict__ out) {
    __shared__ float red[E_DIM];
    const int e = threadIdx.x;
    const float f = Cnt[e] * (1.0f / (float)(B_DIM * TOPK));
    const float P = Psum[e] * (1.0f / (float)B_DIM);
    red[e] = f * P;
    __syncthreads();
    for (int s = E_DIM / 2; s > 0; s >>= 1) {
        if (e < s) red[e] += red[e + s];
        __syncthreads();
    }
    if (e == 0) out[2 * TOPK * B_DIM] = (float)E_DIM * red[0];
}

extern "C" void kernel_launch(void* const* d_in, const int* in_sizes, int n_in,
                              void* d_out, int out_size, void* d_ws, size_t ws_size,
                              hipStream_t stream) {
    const float* A = (const float*)d_in[0];   // [16384, 4096]
    const float* W = (const float*)d_in[1];   // [64, 4096]
    float* out  = (float*)d_out;              // [2B weights | 2B indices | aux]
    float* Psum = (float*)d_ws;               // 64 floats
    float* Cnt  = Psum + E_DIM;               // 64 floats

    router_ws_zero<<<1, 128, 0, stream>>>(Psum);
    router_gemm_topk<<<B_DIM / ROWS_PER_BLOCK, WAVES_PER_BLOCK * 32, 0, stream>>>(
        A, W, out, Psum, Cnt);
    router_finalize<<<1, E_DIM, 0, stream>>>(Psum, Cnt, out);
}